// KGEncoder_30751965839789
// MI455X (gfx1250) — compile-verified
//
#include <hip/hip_runtime.h>
#include <hip/hip_bf16.h>
#include <math.h>

// CDNA5 / gfx1250: wave32, WMMA fp32 16x16x4.
typedef float v2f __attribute__((ext_vector_type(2)));
typedef float v8f __attribute__((ext_vector_type(8)));

#define D_FEAT 128
#define ROWPAD 132   // LDS row stride in dwords; 132 % 64 == 4 -> conflict-free b64 operand loads
#define WAVES_PER_BLOCK 4
#define BLOCK_THREADS (WAVES_PER_BLOCK * 32)

// ---------------------------------------------------------------------------
// Edge kernel: for each 16-edge tile (one wave per tile):
//   comp_h = rotate(ent[src], rel[etype]*norm)          (staged in LDS)
//   accO   = comp_h @ WO.T ;  accI = comp_h @ WI.T      (fp32 WMMA 16x16x4)
//   per-row select (in_mask ? I : out_mask ? O : 0) + bias, atomicAdd to out[dst].
// Post-GEMM selection keeps the WMMA inner loops free of extra VALU (fp32 WMMA
// does not co-execute with VALU on CDNA5).
// ---------------------------------------------------------------------------
__global__ __launch_bounds__(BLOCK_THREADS) void kg_edge_kernel(
    const float* __restrict__ ent, const float* __restrict__ rel,
    const float* __restrict__ norm,
    const int* __restrict__ src, const int* __restrict__ dstI,
    const int* __restrict__ etype,
    const unsigned char* __restrict__ in_mask,
    const unsigned char* __restrict__ out_mask,
    const float* __restrict__ WO, const float* __restrict__ WOb,
    const float* __restrict__ WI, const float* __restrict__ WIb,
    float* __restrict__ out, int E)
{
    __shared__ __align__(16) float sComp[WAVES_PER_BLOCK][16][ROWPAD];
    __shared__ __align__(16) float sW[16][ROWPAD];

    const int lane = threadIdx.x & 31;
    const int wv   = threadIdx.x >> 5;
    const int nTiles = (E + 15) >> 4;
    const int tile = blockIdx.x * WAVES_PER_BLOCK + wv;
    const bool valid = tile < nTiles;          // wave-uniform
    const int e0 = tile << 4;

    // ---- Phase 1: build comp_h tile (16 x 128) into LDS, coalesced per edge row
    if (valid) {
        for (int i = 0; i < 16; ++i) {
            const int e = e0 + i;
            if (e < E) {
                const int   s  = src[e];
                const int   et = etype[e];
                const float nr = norm[e];
                const float* h  = ent + (size_t)s  * D_FEAT;
                const float* rr = rel + (size_t)et * D_FEAT;
#pragma unroll
                for (int p = 0; p < 2; ++p) {
                    const int jj = lane + p * 32;           // complex index 0..63
                    const float hre = h[jj], him = h[jj + 64];
                    const float rre = rr[jj] * nr, rim = rr[jj + 64] * nr;
                    sComp[wv][i][jj]      = hre * rre - him * rim;
                    sComp[wv][i][jj + 64] = hre * rim + him * rre;
                }
            } else {
                sComp[wv][i][lane]      = 0.f;
                sComp[wv][i][lane + 32] = 0.f;
                sComp[wv][i][lane + 64] = 0.f;
                sComp[wv][i][lane + 96] = 0.f;
            }
        }
    }
    __syncthreads();

    // ---- Phase 2: load A operand (this lane's row, its K-halves) into registers
    const int rowa = lane & 15;                 // A-matrix row owned by this lane
    const int koff = (lane < 16) ? 0 : 2;       // K sub-offset per ISA A layout
    v2f a[32];
    if (valid) {
#pragma unroll
        for (int c = 0; c < 32; ++c)
            a[c] = *(const v2f*)&sComp[wv][rowa][4 * c + koff];
    }

    // per-output-row (C/D layout) metadata: dst row + matrix/bias selector
    const int off8 = (lane < 16) ? 0 : 8;
    int dstv[8];
    int bsel[8];                                // -1 invalid, 0 none, 1 -> WO, 2 -> WI
    if (valid) {
#pragma unroll
        for (int v = 0; v < 8; ++v) {
            const int e = e0 + v + off8;
            if (e < E) {
                dstv[v] = dstI[e];
                const bool mi = in_mask[e] != 0;
                const bool mo = out_mask[e] != 0;
                bsel[v] = mi ? 2 : (mo ? 1 : 0);
            } else {
                dstv[v] = 0;
                bsel[v] = -1;
            }
        }
    }

    // ---- Phase 3: per output column-tile, stream W chunks through LDS and WMMA
    for (int jt = 0; jt < 8; ++jt) {
        v8f accO = {};
        v8f accI = {};
        // --- WO chunk
        __syncthreads();
        {
            const int r  = threadIdx.x >> 3;
            const int c0 = (threadIdx.x & 7) << 4;
            const float4* wsrc = (const float4*)(WO + (size_t)(jt * 16 + r) * D_FEAT + c0);
            float4* wdst = (float4*)&sW[r][c0];
            wdst[0] = wsrc[0]; wdst[1] = wsrc[1]; wdst[2] = wsrc[2]; wdst[3] = wsrc[3];
        }
        __syncthreads();
        if (valid) {
#pragma unroll
            for (int c = 0; c < 32; ++c) {
                const v2f bv = *(const v2f*)&sW[rowa][4 * c + koff];
                accO = __builtin_amdgcn_wmma_f32_16x16x4_f32(false, a[c], false, bv,
                                                             (short)0, accO, false, false);
            }
        }
        // --- WI chunk
        __syncthreads();
        {
            const int r  = threadIdx.x >> 3;
            const int c0 = (threadIdx.x & 7) << 4;
            const float4* wsrc = (const float4*)(WI + (size_t)(jt * 16 + r) * D_FEAT + c0);
            float4* wdst = (float4*)&sW[r][c0];
            wdst[0] = wsrc[0]; wdst[1] = wsrc[1]; wdst[2] = wsrc[2]; wdst[3] = wsrc[3];
        }
        __syncthreads();
        if (valid) {
#pragma unroll
            for (int c = 0; c < 32; ++c) {
                const v2f bv = *(const v2f*)&sW[rowa][4 * c + koff];
                accI = __builtin_amdgcn_wmma_f32_16x16x4_f32(false, a[c], false, bv,
                                                             (short)0, accI, false, false);
            }
            // --- per-row select + bias, scatter-add this 16x16 result tile
            const int col = (jt << 4) + rowa;
            const float bO = WOb[col];
            const float bI = WIb[col];
#pragma unroll
            for (int v = 0; v < 8; ++v) {
                if (bsel[v] > 0) {
                    const float val = (bsel[v] == 2) ? (accI[v] + bI) : (accO[v] + bO);
                    atomicAdd(&out[(size_t)dstv[v] * D_FEAT + col], val);
                }
            }
        }
    }
}

// ---------------------------------------------------------------------------
// Self-loop kernel: out[n] = rotate(ent[n], loop_rel) @ WS.T + WS_b
// (initializes the accumulation buffer; edge kernel atomically adds on top)
// ---------------------------------------------------------------------------
__global__ __launch_bounds__(BLOCK_THREADS) void kg_self_kernel(
    const float* __restrict__ ent, const float* __restrict__ loop_rel,
    const float* __restrict__ WS, const float* __restrict__ WSb,
    float* __restrict__ out, int N)
{
    __shared__ __align__(16) float sComp[WAVES_PER_BLOCK][16][ROWPAD];
    __shared__ __align__(16) float sW[16][ROWPAD];

    const int lane = threadIdx.x & 31;
    const int wv   = threadIdx.x >> 5;
    const int nTiles = (N + 15) >> 4;
    const int tile = blockIdx.x * WAVES_PER_BLOCK + wv;
    const bool valid = tile < nTiles;
    const int n0 = tile << 4;

    if (valid) {
        for (int i = 0; i < 16; ++i) {
            const int n = n0 + i;
            if (n < N) {
                const float* h = ent + (size_t)n * D_FEAT;
#pragma unroll
                for (int p = 0; p < 2; ++p) {
                    const int jj = lane + p * 32;
                    const float hre = h[jj], him = h[jj + 64];
                    const float rre = loop_rel[jj], rim = loop_rel[jj + 64];
                    sComp[wv][i][jj]      = hre * rre - him * rim;
                    sComp[wv][i][jj + 64] = hre * rim + him * rre;
                }
            } else {
                sComp[wv][i][lane]      = 0.f;
                sComp[wv][i][lane + 32] = 0.f;
                sComp[wv][i][lane + 64] = 0.f;
                sComp[wv][i][lane + 96] = 0.f;
            }
        }
    }
    __syncthreads();

    const int rowa = lane & 15;
    const int koff = (lane < 16) ? 0 : 2;
    v2f a[32];
    if (valid) {
#pragma unroll
        for (int c = 0; c < 32; ++c)
            a[c] = *(const v2f*)&sComp[wv][rowa][4 * c + koff];
    }
    const int off8 = (lane < 16) ? 0 : 8;

    for (int jt = 0; jt < 8; ++jt) {
        __syncthreads();
        {
            const int r  = threadIdx.x >> 3;
            const int c0 = (threadIdx.x & 7) << 4;
            const float4* wsrc = (const float4*)(WS + (size_t)(jt * 16 + r) * D_FEAT + c0);
            float4* wdst = (float4*)&sW[r][c0];
            wdst[0] = wsrc[0]; wdst[1] = wsrc[1]; wdst[2] = wsrc[2]; wdst[3] = wsrc[3];
        }
        __syncthreads();
        if (valid) {
            v8f acl = {};
#pragma unroll
            for (int c = 0; c < 32; ++c) {
                const v2f bv = *(const v2f*)&sW[rowa][4 * c + koff];
                acl = __builtin_amdgcn_wmma_f32_16x16x4_f32(false, a[c], false, bv,
                                                            (short)0, acl, false, false);
            }
            const int col = (jt << 4) + rowa;
            const float b = WSb[col];
#pragma unroll
            for (int v = 0; v < 8; ++v) {
                const int n = n0 + v + off8;
                if (n < N) out[(size_t)n * D_FEAT + col] = acl[v] + b;
            }
        }
    }
}

// ---------------------------------------------------------------------------
// BN statistics: per-column sum / sumsq of x/3 (block partials + f32 atomics)
// ---------------------------------------------------------------------------
__global__ __launch_bounds__(256) void kg_stats_kernel(
    const float* __restrict__ x, float* __restrict__ stats, int N)
{
    const int col = threadIdx.x & 127;
    const int sub = threadIdx.x >> 7;   // 0..1
    float s = 0.f, s2 = 0.f;
    for (int r = blockIdx.x * 2 + sub; r < N; r += gridDim.x * 2) {
        const float v = x[(size_t)r * D_FEAT + col] * (1.0f / 3.0f);
        s += v; s2 += v * v;
    }
    __shared__ float sh[256], sh2[256];
    sh[threadIdx.x] = s; sh2[threadIdx.x] = s2;
    __syncthreads();
    if (sub == 0) {
        atomicAdd(&stats[col],       s  + sh[col + 128]);
        atomicAdd(&stats[128 + col], s2 + sh2[col + 128]);
    }
}

// ---------------------------------------------------------------------------
// In-place BN + tanh: y = tanh(((x/3)-mu)*rsqrt(var+eps)*gamma + beta)
// ---------------------------------------------------------------------------
__global__ __launch_bounds__(256) void kg_bn_kernel(
    float* __restrict__ x, const float* __restrict__ stats,
    const float* __restrict__ gamma, const float* __restrict__ beta, int N)
{
    const long long idx = (long long)blockIdx.x * blockDim.x + threadIdx.x;
    if (idx >= (long long)N * D_FEAT) return;
    const int col = (int)(idx & 127);
    const float invN = 1.0f / (float)N;
    const float mu  = stats[col] * invN;
    const float var = stats[128 + col] * invN - mu * mu;
    const float v = x[idx] * (1.0f / 3.0f);
    const float y = (v - mu) * rsqrtf(var + 1e-5f) * gamma[col] + beta[col];
    x[idx] = tanhf(y);
}

// ---------------------------------------------------------------------------
// rel_out[r] = rel[r] @ WR.T + WR_b  (tiny: 200x128x128)
// ---------------------------------------------------------------------------
__global__ __launch_bounds__(128) void kg_rel_kernel(
    const float* __restrict__ rel, const float* __restrict__ WR,
    const float* __restrict__ WRb, float* __restrict__ out, int R)
{
    const int r = blockIdx.x;
    const int j = threadIdx.x;
    __shared__ float sr[128];
    sr[j] = rel[(size_t)r * D_FEAT + j];
    __syncthreads();
    float s = 0.f;
    const float* wrow = WR + (size_t)j * D_FEAT;
#pragma unroll 8
    for (int k = 0; k < D_FEAT; ++k) s += sr[k] * wrow[k];
    out[(size_t)r * D_FEAT + j] = s + WRb[j];
}

// ---------------------------------------------------------------------------
extern "C" void kernel_launch(void* const* d_in, const int* in_sizes, int n_in,
                              void* d_out, int out_size, void* d_ws, size_t ws_size,
                              hipStream_t stream) {
    (void)n_in; (void)out_size; (void)ws_size;
    const float* ent      = (const float*)d_in[0];
    const float* rel      = (const float*)d_in[1];
    const float* norm     = (const float*)d_in[2];
    const float* loop_rel = (const float*)d_in[3];
    const float* WO_w = (const float*)d_in[4];
    const float* WO_b = (const float*)d_in[5];
    const float* WI_w = (const float*)d_in[6];
    const float* WI_b = (const float*)d_in[7];
    const float* WS_w = (const float*)d_in[8];
    const float* WS_b = (const float*)d_in[9];
    const float* WR_w = (const float*)d_in[10];
    const float* WR_b = (const float*)d_in[11];
    const float* gamma = (const float*)d_in[12];
    const float* beta  = (const float*)d_in[13];
    const int* src   = (const int*)d_in[14];
    const int* dst   = (const int*)d_in[15];
    const int* etype = (const int*)d_in[16];
    const unsigned char* in_mask  = (const unsigned char*)d_in[17];  // numpy bool (1 byte)
    const unsigned char* out_mask = (const unsigned char*)d_in[18];

    const int N = in_sizes[0] / D_FEAT;
    const int R = in_sizes[1] / D_FEAT;
    const int E = in_sizes[2];

    float* entOut = (float*)d_out;                       // N*128, used as accumulator then BN'd in place
    float* relOut = entOut + (size_t)N * D_FEAT;         // R*128
    float* stats  = (float*)d_ws;                        // 256 floats

    hipMemsetAsync(stats, 0, 256 * sizeof(float), stream);

    const int tilesN = (N + 15) / 16;
    kg_self_kernel<<<(tilesN + WAVES_PER_BLOCK - 1) / WAVES_PER_BLOCK, BLOCK_THREADS, 0, stream>>>(
        ent, loop_rel, WS_w, WS_b, entOut, N);

    const int tilesE = (E + 15) / 16;
    kg_edge_kernel<<<(tilesE + WAVES_PER_BLOCK - 1) / WAVES_PER_BLOCK, BLOCK_THREADS, 0, stream>>>(
        ent, rel, norm, src, dst, etype, in_mask, out_mask,
        WO_w, WO_b, WI_w, WI_b, entOut, E);

    kg_stats_kernel<<<512, 256, 0, stream>>>(entOut, stats, N);

    const long long tot = (long long)N * D_FEAT;
    kg_bn_kernel<<<(unsigned)((tot + 255) / 256), 256, 0, stream>>>(entOut, stats, gamma, beta, N);

    kg_rel_kernel<<<R, 128, 0, stream>>>(rel, WR_w, WR_b, relOut, R);
}